// GRUStack_46007689675059
// MI455X (gfx1250) — compile-verified
//
#include <hip/hip_runtime.h>

#define T_STEPS 8192
#define HDIM    1024
#define NGATES  3072   // 3*H

typedef float v2f __attribute__((ext_vector_type(2)));
typedef float v8f __attribute__((ext_vector_type(8)));
typedef unsigned int u32x4 __attribute__((ext_vector_type(4)));
typedef int i32x4 __attribute__((ext_vector_type(4)));
typedef int i32x8 __attribute__((ext_vector_type(8)));

#if __has_builtin(__builtin_amdgcn_tensor_load_to_lds) && \
    __has_builtin(__builtin_amdgcn_s_wait_tensorcnt)
#define HAVE_TDM 1
#else
#define HAVE_TDM 0
#endif

// ---------------------------------------------------------------------------
// Phase 1: igates = X @ Wih^T + b   (T x K) @ (K x 3H) -> (T x 3H)
// fp32 WMMA 16x16x4, 64x64 WG tile, BK=32, 8 waves (2x4), 2 M-subtiles/wave.
// Tile movement: TDM tensor_load_to_lds (double-buffered, padded rows) when
// available; synchronous global->LDS copy otherwise.
// ---------------------------------------------------------------------------
#define GM 64
#define GN 64
#define GK 32
#define GKP (GK + 4)   // 36-DWORD LDS row stride: b64/b128 aligned, banks spread

#if HAVE_TDM
// 2-D TDM descriptor: 64 rows x 32 f32, row stride = K elems, LDS pad 4 DW / 32 DW
__device__ __forceinline__ void tdm_load_tile(const float* gsrc,
                                              unsigned lds_byte_off, int K)
{
  unsigned long long ga = (unsigned long long)(uintptr_t)gsrc;
  u32x4 g0;
  g0.x = 1u;                                          // count=1 (valid), user mode
  g0.y = lds_byte_off;                                // lds_addr [63:32]
  g0.z = (unsigned)(ga & 0xffffffffu);                // global_addr [95:64]
  g0.w = (unsigned)((ga >> 32) & 0x01ffffffu)         // global_addr [120:96]
       | (2u << 30);                                  // type=2 ("image")
  i32x8 g1;
  g1[0] = (2 << 16)            // data_size = 4 bytes
        | (1 << 20)            // pad_enable
        | (4 << 22)            // pad_interval: 2^(4+1)=32 DWORDs per row
        | (3 << 25);           // pad_amount: 3+1 = 4 DWORDs  -> row stride 36 DW
  g1[1] = (int)(1024u << 16);  // tensor_dim0 = 1024 (elems), low 16 bits
  g1[2] = (int)(0xFFFFu << 16);// tensor_dim0 hi=0 | tensor_dim1 lo16 = 65535
  g1[3] = (int)(32u << 16);    // tensor_dim1 hi=0 | tile_dim0 = 32
  g1[4] = 64;                  // tile_dim1 = 64 rows | tile_dim2 = 0
  g1[5] = K;                   // tensor_dim0_stride low 32 (elems)
  g1[6] = 0;                   // stride hi | tensor_dim1_stride lo
  g1[7] = 0;
  i32x4 z4 = {0, 0, 0, 0};
#if __clang_major__ >= 23
  i32x8 z8 = {0, 0, 0, 0, 0, 0, 0, 0};
  __builtin_amdgcn_tensor_load_to_lds(g0, g1, z4, z4, z8, 0);
#else
  __builtin_amdgcn_tensor_load_to_lds(g0, g1, z4, z4, 0);
#endif
}
#endif

__global__ __launch_bounds__(256) void gemm_igates(
    const float* __restrict__ X,     // T x K row-major
    const float* __restrict__ W,     // 3H x K row-major (Wih)
    const float* __restrict__ bias,  // 3H
    float* __restrict__ out,         // T x 3H
    int K, int N)
{
  __shared__ float As[2][GM][GKP];
  __shared__ float Bs[2][GN][GKP];
  const int tid  = threadIdx.x;
  const int lane = tid & 31;
  const int wave = tid >> 5;
  const int wm   = wave >> 2;      // 0..1
  const int wn   = wave & 3;       // 0..3
  const int m0   = blockIdx.y * GM;
  const int n0   = blockIdx.x * GN;
  const int half = lane >> 4;      // 0 or 1
  const int l16  = lane & 15;

  v8f acc0 = {0.f,0.f,0.f,0.f,0.f,0.f,0.f,0.f};
  v8f acc1 = {0.f,0.f,0.f,0.f,0.f,0.f,0.f,0.f};

#if HAVE_TDM
  // ---------------- TDM double-buffered pipeline ----------------
  const unsigned offA[2] = { (unsigned)(uintptr_t)&As[0][0][0],
                             (unsigned)(uintptr_t)&As[1][0][0] };
  const unsigned offB[2] = { (unsigned)(uintptr_t)&Bs[0][0][0],
                             (unsigned)(uintptr_t)&Bs[1][0][0] };
  if (wave == 0) {
    tdm_load_tile(X + (size_t)m0 * K, offA[0], K);
    tdm_load_tile(W + (size_t)n0 * K, offB[0], K);
  }
  int buf = 0;
  for (int k0 = 0; k0 < K; k0 += GK) {
    if (wave == 0) __builtin_amdgcn_s_wait_tensorcnt(0);
    __syncthreads();                       // tile[buf] visible to all waves
    const int nbuf = buf ^ 1;
    if (wave == 0 && k0 + GK < K) {        // prefetch next stage via TDM
      tdm_load_tile(X + (size_t)m0 * K + k0 + GK, offA[nbuf], K);
      tdm_load_tile(W + (size_t)n0 * K + k0 + GK, offB[nbuf], K);
    }
    #pragma unroll
    for (int ks = 0; ks < GK; ks += 4) {
      v2f bfrag = *(const v2f*)&Bs[buf][wn * 16 + l16][ks + 2 * half];
      v2f a0    = *(const v2f*)&As[buf][wm * 32 +      l16][ks + 2 * half];
      v2f a1    = *(const v2f*)&As[buf][wm * 32 + 16 + l16][ks + 2 * half];
      acc0 = __builtin_amdgcn_wmma_f32_16x16x4_f32(false, a0, false, bfrag,
                                                   (short)0, acc0, false, false);
      acc1 = __builtin_amdgcn_wmma_f32_16x16x4_f32(false, a1, false, bfrag,
                                                   (short)0, acc1, false, false);
    }
    __syncthreads();                       // all reads of tile[buf] done
    buf = nbuf;
  }
#else
  // ---------------- synchronous fallback ----------------
  for (int k0 = 0; k0 < K; k0 += GK) {
    #pragma unroll
    for (int u = 0; u < 2; ++u) {
      int idx = tid * 2 + u;
      int r   = idx >> 3;
      int c   = (idx & 7) << 2;
      float4 va = *(const float4*)(X + (size_t)(m0 + r) * K + k0 + c);
      As[0][r][c+0] = va.x; As[0][r][c+1] = va.y;
      As[0][r][c+2] = va.z; As[0][r][c+3] = va.w;
      float4 vb = *(const float4*)(W + (size_t)(n0 + r) * K + k0 + c);
      Bs[0][r][c+0] = vb.x; Bs[0][r][c+1] = vb.y;
      Bs[0][r][c+2] = vb.z; Bs[0][r][c+3] = vb.w;
    }
    __syncthreads();
    #pragma unroll
    for (int ks = 0; ks < GK; ks += 4) {
      v2f bfrag = *(const v2f*)&Bs[0][wn * 16 + l16][ks + 2 * half];
      v2f a0    = *(const v2f*)&As[0][wm * 32 +      l16][ks + 2 * half];
      v2f a1    = *(const v2f*)&As[0][wm * 32 + 16 + l16][ks + 2 * half];
      acc0 = __builtin_amdgcn_wmma_f32_16x16x4_f32(false, a0, false, bfrag,
                                                   (short)0, acc0, false, false);
      acc1 = __builtin_amdgcn_wmma_f32_16x16x4_f32(false, a1, false, bfrag,
                                                   (short)0, acc1, false, false);
    }
    __syncthreads();
  }
#endif

  // C/D layout: VGPR v -> M = v + 8*half, N = l16
  const int   col = n0 + wn * 16 + l16;
  const float bv  = bias[col];
  #pragma unroll
  for (int v = 0; v < 8; ++v) {
    int ra = m0 + wm * 32 + v + 8 * half;
    out[(size_t)ra * N + col]        = acc0[v] + bv;
    out[(size_t)(ra + 16) * N + col] = acc1[v] + bv;
  }
}

// ---------------------------------------------------------------------------
// Phase 2: persistent sequential GRU scan.
// 64 WGs x 256 threads. WG w owns outputs j in [w*16, w*16+16) and the 48
// corresponding rows of Whh (r, z, n gates), pinned in VGPRs: each lane holds
// 6 rows x 16 float2 = 192 floats for the whole 8192-step scan.
// Per step: h_{t-1} -> LDS, register-resident dot products, wave32 xor-reduce,
// gate nonlinearities, device-wide sense-reversing barrier.
// ---------------------------------------------------------------------------
#define SCAN_WGS       64
#define OUT_PER_WG     16
#define ROWS_PER_WG    48
#define ROWS_PER_WAVE  6

struct GridBarrier { unsigned count; unsigned gen; };

__global__ void init_barrier(GridBarrier* b) { b->count = 0u; b->gen = 0u; }

__global__ __launch_bounds__(256, 1) void gru_scan(
    const float* __restrict__ igates,  // T x 3H
    const float* __restrict__ Whh,     // 3H x H
    const float* __restrict__ bn,      // H
    float* __restrict__ out,           // T x H (also h history)
    GridBarrier* __restrict__ bar)
{
  __shared__ float hsh[HDIM];
  __shared__ float hg[ROWS_PER_WG];

  const int tid  = threadIdx.x;
  const int wg   = blockIdx.x;
  const int wave = tid >> 5;
  const int lane = tid & 31;

  // ---- pin this wave's 6 Whh rows in VGPRs (each lane: pair index lane+32*i)
  float2 wreg[ROWS_PER_WAVE][16];
  #pragma unroll
  for (int q = 0; q < ROWS_PER_WAVE; ++q) {
    int r = wave * ROWS_PER_WAVE + q;        // 0..47
    int g = r >> 4, o = r & 15;              // gate, output-within-WG
    const float2* wrow =
        (const float2*)(Whh + (size_t)(g * HDIM + wg * OUT_PER_WG + o) * HDIM);
    #pragma unroll
    for (int i = 0; i < 16; ++i) wreg[q][i] = wrow[lane + 32 * i];
  }

  for (int t = 0; t < T_STEPS; ++t) {
    // h_{t-1} into LDS (zeros at t==0)
    float4* h4 = (float4*)hsh;
    if (t == 0) {
      for (int i = tid; i < HDIM / 4; i += 256)
        h4[i] = make_float4(0.f, 0.f, 0.f, 0.f);
    } else {
      const float4* hp = (const float4*)(out + (size_t)(t - 1) * HDIM);
      for (int i = tid; i < HDIM / 4; i += 256) h4[i] = hp[i];
    }
    __syncthreads();

    // hoist this lane's h pairs (bank-conflict-free ds_load_b64)
    float2 hreg[16];
    const float2* h2 = (const float2*)hsh;
    #pragma unroll
    for (int i = 0; i < 16; ++i) hreg[i] = h2[lane + 32 * i];

    // register-resident dot products + wave32 reduction
    #pragma unroll
    for (int q = 0; q < ROWS_PER_WAVE; ++q) {
      float acc = 0.f;
      #pragma unroll
      for (int i = 0; i < 16; ++i)
        acc = fmaf(wreg[q][i].x, hreg[i].x,
              fmaf(wreg[q][i].y, hreg[i].y, acc));
      acc += __shfl_xor(acc, 16);
      acc += __shfl_xor(acc, 8);
      acc += __shfl_xor(acc, 4);
      acc += __shfl_xor(acc, 2);
      acc += __shfl_xor(acc, 1);
      if (lane == 0) hg[wave * ROWS_PER_WAVE + q] = acc;
    }
    __syncthreads();

    // gates + output for the 16 owned h elements
    if (tid < OUT_PER_WG) {
      const int o = tid;
      const int j = wg * OUT_PER_WG + o;
      const size_t base = (size_t)t * NGATES;
      float igr = igates[base + j];
      float igz = igates[base + HDIM + j];
      float ign = igates[base + 2 * HDIM + j];
      float hr = hg[o], hz = hg[16 + o], hn = hg[32 + o];
      float r = 1.f / (1.f + __expf(-(igr + hr)));
      float z = 1.f / (1.f + __expf(-(igz + hz)));
      float n = tanhf(ign + r * (hn + bn[j]));
      float hprev = hsh[j];
      out[(size_t)t * HDIM + j] = n + z * (hprev - n);
      if (t + 1 < T_STEPS)
        __builtin_prefetch(igates + base + NGATES + j, 0, 1);
    }
    __threadfence();
    __syncthreads();

    // device-wide sense-reversing barrier (one lane per WG)
    if (tid == 0) {
      unsigned g0 = __hip_atomic_load(&bar->gen, __ATOMIC_RELAXED,
                                      __HIP_MEMORY_SCOPE_AGENT);
      unsigned my = __hip_atomic_fetch_add(&bar->count, 1u, __ATOMIC_ACQ_REL,
                                           __HIP_MEMORY_SCOPE_AGENT);
      if (my == SCAN_WGS - 1) {
        __hip_atomic_store(&bar->count, 0u, __ATOMIC_RELAXED,
                           __HIP_MEMORY_SCOPE_AGENT);
        __hip_atomic_fetch_add(&bar->gen, 1u, __ATOMIC_RELEASE,
                               __HIP_MEMORY_SCOPE_AGENT);
      } else {
        while (__hip_atomic_load(&bar->gen, __ATOMIC_ACQUIRE,
                                 __HIP_MEMORY_SCOPE_AGENT) == g0) {
          __builtin_amdgcn_s_sleep(1);
        }
      }
    }
    __syncthreads();
  }
}

// ---------------------------------------------------------------------------
// Host side: 4 layers of (WMMA GEMM -> barrier init -> persistent scan)
// ws layout: igates (96 MB) | xbuf0 (32 MB) | xbuf1 (32 MB) | GridBarrier
// ---------------------------------------------------------------------------
extern "C" void kernel_launch(void* const* d_in, const int* in_sizes, int n_in,
                              void* d_out, int out_size, void* d_ws, size_t ws_size,
                              hipStream_t stream) {
  (void)in_sizes; (void)n_in; (void)out_size; (void)ws_size;
  const int K = HDIM;  // D == H == 1024 for every layer

  float* igates = (float*)d_ws;
  float* xbuf0  = igates + (size_t)T_STEPS * NGATES;
  float* xbuf1  = xbuf0  + (size_t)T_STEPS * HDIM;
  GridBarrier* bar = (GridBarrier*)(xbuf1 + (size_t)T_STEPS * HDIM);

  const float* xcur = (const float*)d_in[0];
  for (int l = 0; l < 4; ++l) {
    const float* Wih = (const float*)d_in[1 + 4 * l];
    const float* Whh = (const float*)d_in[2 + 4 * l];
    const float* b   = (const float*)d_in[3 + 4 * l];
    const float* bn  = (const float*)d_in[4 + 4 * l];

    dim3 gg(NGATES / GN, T_STEPS / GM);
    gemm_igates<<<gg, 256, 0, stream>>>(xcur, Wih, b, igates, K, NGATES);

    float* xout = (l == 3) ? (float*)d_out : ((l & 1) ? xbuf1 : xbuf0);
    init_barrier<<<1, 1, 0, stream>>>(bar);
    gru_scan<<<SCAN_WGS, 256, 0, stream>>>(igates, Whh, bn, xout, bar);
    xcur = xout;
  }
}